// SchroederReverb_52630529245323
// MI455X (gfx1250) — compile-verified
//
#include <hip/hip_runtime.h>

// Schroeder reverb for MI455X (gfx1250, wave32).
//
// Each stage is y[b] = a*y[b-1] + u[b] on N decoupled stride-N subsequences
// (per channel). We chunk the scan into groups of 16 time-blocks and compute
//   Y(16 x 16ch) = L(16x16) @ U(16 x 16ch) + outer(a^(i+1), carry)
// with L[i][j] = a^(i-j) (i>=j): a 16x16x16 f32 GEMM = 4 chained
// V_WMMA_F32_16X16X4_F32 ops, the native CDNA5 f32 matrix path.
//
// N is a template parameter so all strides/divisions fold to constants and
// the hot loop is pure { const-offset global_load_b32 -> v_wmma -> store }.
// Only chunk 0 (forced-zero block) and the partial tail use (branch-free,
// cndmask-based) guards, keeping EXEC all-ones around every WMMA.
//
// Needs ws_size >= T*W*4 bytes (64 MB) for one ping-pong buffer.

typedef float v2f __attribute__((ext_vector_type(2)));
typedef float v8f __attribute__((ext_vector_type(8)));

#define T_LEN 131072
#define W_CH  128

__device__ __forceinline__ float powi_pos(float a, int e) {
  // a^e for e >= 0, else 0 (triangular fill).
  if (e < 0) return 0.0f;
  float r = 1.0f;
  for (int i = 0; i < e; ++i) r *= a;
  return r;
}

// ALLPASS: u[b] = x[b] + g*x[b-1], a = -g   (b >= 1; u[0] = 0 per reference)
// comb:    u[b] = x[b],            a = +g   (b >= 1; u[0] = 0 per reference)
// ACCUM:   out[...] += y  instead of  out[...] = y
template <int N, bool ALLPASS, bool ACCUM>
__global__ __launch_bounds__(256) void filter_stage(const float* __restrict__ in,
                                                    float* __restrict__ out,
                                                    float a, float g) {
  constexpr long NW = (long)N * W_CH;        // row stride between time-blocks
  const int lane = threadIdx.x & 31;
  const int wave = threadIdx.x >> 5;         // 0..7 : channel tile
  const int r    = blockIdx.x;               // residue 0..N-1
  const int col  = lane & 15;                // row (A) / column (B,C,D) index
  const int hf   = lane >> 4;                // wave half
  const int c    = (wave << 4) + col;        // channel 0..127

  // A operand: 16x4 K-slices of L per the ISA layout
  // (lanes 0-15: M=lane, V0=K0 V1=K1; lanes 16-31: M=lane-16, V0=K2 V1=K3).
  v2f A[4];
#pragma unroll
  for (int k = 0; k < 4; ++k) {
    const int j0 = 4 * k + 2 * hf;
    A[k].x = powi_pos(a, col - j0);          // L[m][j0]
    A[k].y = powi_pos(a, col - j0 - 1);      // L[m][j0+1]
  }
  // C seed factors: C[v] holds row i = v + 8*hf -> a^(i+1) * carry.
  float cf[8];
#pragma unroll
  for (int v = 0; v < 8; ++v) cf[v] = powi_pos(a, v + 8 * hf + 1);

  const int maxb   = (T_LEN - 1 - r) / N;    // last block with t < T (const-div)
  const int chunks = maxb / 16 + 1;
  const int full   = (maxb - 15) / 16 + 1;   // chunks with all 16 rows in range

  float carry = 0.0f;

  // ---- guarded chunk: branch-free cndmask guards (chunk 0 and tail) ----
  auto chunk_guarded = [&](int q) {
    const int b0 = q * 16;
    v8f Cm;
#pragma unroll
    for (int v = 0; v < 8; ++v) Cm[v] = cf[v] * carry;
#pragma unroll
    for (int k = 0; k < 4; ++k) {
      const int i0 = 4 * k + 2 * hf;
      v2f Bk;
#pragma unroll
      for (int e = 0; e < 2; ++e) {
        const int  b  = b0 + i0 + e;
        const long t  = (long)b * N + r;
        const bool ok = (b >= 1) && (t < T_LEN);
        const long tc = ok ? t : 0;          // clamped address: always valid
        float u = in[tc * W_CH + c];
        if (ALLPASS) {
          const long tm = ok ? (t - N) : 0;  // b>=1 -> t-N >= 0
          u += g * in[tm * W_CH + c];
        }
        u = ok ? u : 0.0f;
        if (e == 0) Bk.x = u; else Bk.y = u;
      }
      Cm = __builtin_amdgcn_wmma_f32_16x16x4_f32(false, A[k], false, Bk,
                                                 (short)0, Cm, false, false);
    }
#pragma unroll
    for (int v = 0; v < 8; ++v) {
      const long t = ((long)b0 + v + 8 * hf) * N + r;
      if (t < T_LEN) {                       // exec-masked store, tail only
        const size_t idx = (size_t)t * W_CH + c;
        float val = Cm[v];
        if (ACCUM) val += out[idx];
        out[idx] = val;
      }
    }
    carry = __shfl(Cm[7], 16 + col, 32);     // row 15 lives in lanes 16-31, V7
  };

  // ---- fast chunk: b0 >= 16 and all rows in range; const-offset addressing ----
  auto chunk_fast = [&](int q) {
    const long base = ((long)q * 16 * N + r) * W_CH + c;
    const float* p = in + base;              // x row i of chunk at p[i*NW]
    float*       s = out + base;
    v8f Cm;
#pragma unroll
    for (int v = 0; v < 8; ++v) Cm[v] = cf[v] * carry;
#pragma unroll
    for (int k = 0; k < 4; ++k) {
      const int i0 = 4 * k + 2 * hf;
      const float x0 = p[(long)(i0    ) * NW];
      const float x1 = p[(long)(i0 + 1) * NW];
      v2f Bk;
      if (ALLPASS) {
        const float xm = p[(long)(i0 - 1) * NW];  // row -1 = prev chunk, valid
        Bk.x = x0 + g * xm;
        Bk.y = x1 + g * x0;                       // shared x0 load
      } else {
        Bk.x = x0;
        Bk.y = x1;
      }
      Cm = __builtin_amdgcn_wmma_f32_16x16x4_f32(false, A[k], false, Bk,
                                                 (short)0, Cm, false, false);
    }
#pragma unroll
    for (int v = 0; v < 8; ++v) {
      const long off = (long)(v + 8 * hf) * NW;
      float val = Cm[v];
      if (ACCUM) val += s[off];
      s[off] = val;
    }
    carry = __shfl(Cm[7], 16 + col, 32);
  };

  chunk_guarded(0);                                   // contains the b==0 row
  for (int q = 1; q < full; ++q) chunk_fast(q);       // hot loop
  for (int q = (full > 1 ? full : 1); q < chunks; ++q) chunk_guarded(q);  // tail
}

extern "C" void kernel_launch(void* const* d_in, const int* in_sizes, int n_in,
                              void* d_out, int out_size, void* d_ws, size_t ws_size,
                              hipStream_t stream) {
  const float* x = (const float*)d_in[0];
  float* out     = (float*)d_out;
  float* ws0     = (float*)d_ws;       // T*W floats = 64 MB ping-pong buffer

  const dim3 block(256);               // 8 waves: one per 16-channel tile
  const float G = 0.7f;

  // --- serial allpass chain: x -> out -> ws0 -> out -> ws0 ---
  filter_stage<225, true, false><<<225, block, 0, stream>>>(x,   out, -G, G);
  filter_stage<556, true, false><<<556, block, 0, stream>>>(out, ws0, -G, G);
  filter_stage<441, true, false><<<441, block, 0, stream>>>(ws0, out, -G, G);
  filter_stage<341, true, false><<<341, block, 0, stream>>>(out, ws0, -G, G);

  // --- parallel comb bank, summed into d_out (stream-ordered, RMW safe) ---
  filter_stage<1116, false, false><<<1116, block, 0, stream>>>(ws0, out, 0.84f, 0.0f);
  filter_stage<1188, false, true ><<<1188, block, 0, stream>>>(ws0, out, 0.82f, 0.0f);
  filter_stage<1277, false, true ><<<1277, block, 0, stream>>>(ws0, out, 0.80f, 0.0f);
  filter_stage<1356, false, true ><<<1356, block, 0, stream>>>(ws0, out, 0.78f, 0.0f);
}